// PEPS_Model_36721970380866
// MI455X (gfx1250) — compile-verified
//
#include <hip/hip_runtime.h>

// PEPS 6x6, D=4, phys=2, batch=1024.
// Column-by-column exact contraction; every site absorption is a batch of
// 16x16 (K=16) GEMMs executed with V_WMMA_F32_16X16X4_F32 (fp32 matrix pipe).
// 2 samples / 256-thread workgroup, 256KB LDS ping-pong state per block.

typedef float v2f __attribute__((ext_vector_type(2)));
typedef float v8f __attribute__((ext_vector_type(8)));

#define LXR 6
#define LYR 6

__device__ __forceinline__ v8f wmma_f32(v2f a, v2f b, v8f c) {
  // D(16x16) = A(16x4) * B(4x16) + C   (fp32)
  return __builtin_amdgcn_wmma_f32_16x16x4_f32(false, a, false, b, (short)0, c,
                                               false, false);
}

// Generic middle step i in {1,2,3}: out[p,rd,q] = sum_ul in[p,ul,q]*M[ul,rd]
//   GEMM per p: D(16 x Q) = M^T(16x16) @ IN(16 x Q), tiled N by 16.
// Q = 4^(5-i)*? : step1 Q=256, step2 Q=64, step3 Q=16.
template <int Q, int NT_BITS>
__device__ __forceinline__ void step_mid(const float* Tb, const float* IN,
                                         float* OUT, int wsub, int n16, int hi) {
  // A = M^T chunks: lane L: row rd = n16, col k = 4c + 2*hi + {0,1}
  // M[k,rd] = Tb[(k>>2)*64 + (rd>>2)*16 + (rd&3)*4 + (k&3)]  (axes u,r,d,l)
  const int roff = (n16 >> 2) * 16 + (n16 & 3) * 4;
  v2f a[4];
#pragma unroll
  for (int c = 0; c < 4; ++c) {
    const int k0 = 4 * c + 2 * hi;
    a[c].x = Tb[((k0 + 0) >> 2) * 64 + roff + ((k0 + 0) & 3)];
    a[c].y = Tb[((k0 + 1) >> 2) * 64 + roff + ((k0 + 1) & 3)];
  }
  for (int t = wsub * 16; t < wsub * 16 + 16; t += 2) {
    const int p0 = t >> NT_BITS;
    const int p1 = (t + 1) >> NT_BITS;
    const int q0 = (t & ((1 << NT_BITS) - 1)) * 16 + n16;
    const int q1 = ((t + 1) & ((1 << NT_BITS) - 1)) * 16 + n16;
    const float* in0 = IN + p0 * 16 * Q;
    const float* in1 = IN + p1 * 16 * Q;
    v8f c0 = {0.f, 0.f, 0.f, 0.f, 0.f, 0.f, 0.f, 0.f};
    v8f c1 = {0.f, 0.f, 0.f, 0.f, 0.f, 0.f, 0.f, 0.f};
#pragma unroll
    for (int c = 0; c < 4; ++c) {
      const int m0 = 4 * c + 2 * hi;  // B rows: k = m0 + {0,1}
      v2f b0, b1;
      b0.x = in0[(m0 + 0) * Q + q0];
      b0.y = in0[(m0 + 1) * Q + q0];
      b1.x = in1[(m0 + 0) * Q + q1];
      b1.y = in1[(m0 + 1) * Q + q1];
      c0 = wmma_f32(a[c], b0, c0);
      c1 = wmma_f32(a[c], b1, c1);
    }
    float* o0 = OUT + p0 * 16 * Q;
    float* o1 = OUT + p1 * 16 * Q;
#pragma unroll
    for (int v = 0; v < 8; ++v) {
      o0[(v + 8 * hi) * Q + q0] = c0[v];
      o1[(v + 8 * hi) * Q + q1] = c1[v];
    }
  }
}

extern "C" __global__ __launch_bounds__(256) void peps_amp_kernel(
    const int* __restrict__ x, const float* __restrict__ T,
    float* __restrict__ out) {
  __shared__ float smem[65536];  // 2 samples * 2 ping-pong buffers * 16384 f32

  const int tid = threadIdx.x;
  const int lane = tid & 31;
  const int wave = tid >> 5;
  const int sl = wave >> 2;   // local sample (0/1)
  const int wsub = wave & 3;  // wave within sample: owns 16 of 64 tiles
  const int sample = blockIdx.x * 2 + sl;
  const int n16 = lane & 15;
  const int hi = lane >> 4;  // lane half (0/1)

  float* bufA = smem + sl * 32768;
  float* bufB = bufA + 16384;

  // Initial boundary vector over (l0..l5): delta at 0 (dangling left bonds).
  for (int t = (tid & 127); t < 4096; t += 128)
    bufA[t] = (t == 0) ? 1.0f : 0.0f;
  __syncthreads();

  const int xbase = sample * (LXR * LYR);

  for (int jcol = 0; jcol < LYR; ++jcol) {
    // ---- step 0: row 0 (u sliced at 0). IN bufA (4 x 1024) -> OUT bufB (16 x 1024)
    {
      const int spin = x[xbase + 0 * LYR + jcol];
      const float* Tb = T + ((0 * LYR + jcol) * 2 + spin) * 256;
      const int roff = (n16 >> 2) * 16 + (n16 & 3) * 4;  // r,d of output col rd
      v2f a;  // A = M0^T (16x4): M0[l,rd] = Tb[u=0, r, d, l]
      a.x = Tb[roff + (2 * hi + 0)];
      a.y = Tb[roff + (2 * hi + 1)];
      for (int t = wsub * 16; t < wsub * 16 + 16; t += 2) {
        const int q0 = t * 16 + n16;
        const int q1 = q0 + 16;
        const int k0 = 2 * hi;
        v2f b0, b1;
        b0.x = bufA[(k0 + 0) * 1024 + q0];
        b0.y = bufA[(k0 + 1) * 1024 + q0];
        b1.x = bufA[(k0 + 0) * 1024 + q1];
        b1.y = bufA[(k0 + 1) * 1024 + q1];
        v8f c0 = {0.f, 0.f, 0.f, 0.f, 0.f, 0.f, 0.f, 0.f};
        v8f c1 = {0.f, 0.f, 0.f, 0.f, 0.f, 0.f, 0.f, 0.f};
        c0 = wmma_f32(a, b0, c0);
        c1 = wmma_f32(a, b1, c1);
#pragma unroll
        for (int v = 0; v < 8; ++v) {
          bufB[(v + 8 * hi) * 1024 + q0] = c0[v];
          bufB[(v + 8 * hi) * 1024 + q1] = c1[v];
        }
      }
    }
    __syncthreads();

    // ---- steps 1..3: interior rows, GEMM per p, N = Q
    step_mid<256, 4>(T + ((1 * LYR + jcol) * 2 + x[xbase + 1 * LYR + jcol]) * 256,
                     bufB, bufA, wsub, n16, hi);
    __syncthreads();
    step_mid<64, 2>(T + ((2 * LYR + jcol) * 2 + x[xbase + 2 * LYR + jcol]) * 256,
                    bufA, bufB, wsub, n16, hi);
    __syncthreads();
    step_mid<16, 0>(T + ((3 * LYR + jcol) * 2 + x[xbase + 3 * LYR + jcol]) * 256,
                    bufB, bufA, wsub, n16, hi);
    __syncthreads();

    // ---- step 4: P=256,Q=4 -> flip: rows = p, B = M. IN bufA -> OUT bufB
    {
      const int spin = x[xbase + 4 * LYR + jcol];
      const float* Tb = T + ((4 * LYR + jcol) * 2 + spin) * 256;
      const int noff = (n16 >> 2) * 16 + (n16 & 3) * 4;  // output col rd = n16
      v2f bm[4];  // B[k,n] = M[k,n]
#pragma unroll
      for (int c = 0; c < 4; ++c) {
        const int k0 = 4 * c + 2 * hi;
        bm[c].x = Tb[((k0 + 0) >> 2) * 64 + noff + ((k0 + 0) & 3)];
        bm[c].y = Tb[((k0 + 1) >> 2) * 64 + noff + ((k0 + 1) & 3)];
      }
      for (int t = wsub * 16; t < wsub * 16 + 16; t += 2) {
        const int pt0 = t >> 2, q0 = t & 3;
        const int pt1 = (t + 1) >> 2, q1 = (t + 1) & 3;
        const int row0 = pt0 * 16 + n16;
        const int row1 = pt1 * 16 + n16;
        v8f c0 = {0.f, 0.f, 0.f, 0.f, 0.f, 0.f, 0.f, 0.f};
        v8f c1 = {0.f, 0.f, 0.f, 0.f, 0.f, 0.f, 0.f, 0.f};
#pragma unroll
        for (int c = 0; c < 4; ++c) {
          const int m0 = 4 * c + 2 * hi;
          v2f a0, a1;  // A rows = p, cols = k: IN[p*64 + m*4 + q]
          a0.x = bufA[row0 * 64 + (m0 + 0) * 4 + q0];
          a0.y = bufA[row0 * 64 + (m0 + 1) * 4 + q0];
          a1.x = bufA[row1 * 64 + (m0 + 0) * 4 + q1];
          a1.y = bufA[row1 * 64 + (m0 + 1) * 4 + q1];
          c0 = wmma_f32(a0, bm[c], c0);
          c1 = wmma_f32(a1, bm[c], c1);
        }
#pragma unroll
        for (int v = 0; v < 8; ++v) {
          bufB[(pt0 * 16 + v + 8 * hi) * 64 + n16 * 4 + q0] = c0[v];
          bufB[(pt1 * 16 + v + 8 * hi) * 64 + n16 * 4 + q1] = c1[v];
        }
      }
    }
    __syncthreads();

    // ---- step 5: row 5 (d sliced at 0). P=1024,Q=1, B = M5 (16x4, N-padded).
    // IN bufB (1024 x 16) -> OUT bufA (1024 x 4)
    {
      const int spin = x[xbase + 5 * LYR + jcol];
      const float* Tb = T + ((5 * LYR + jcol) * 2 + spin) * 256;
      v2f bm[4];  // B[k,n] = (n<4) ? M5[k, r=n] : 0 ; M5[k,r] = Tb[u,r,0,l]
#pragma unroll
      for (int c = 0; c < 4; ++c) {
        const int k0 = 4 * c + 2 * hi;
        bm[c].x = (n16 < 4) ? Tb[((k0 + 0) >> 2) * 64 + n16 * 16 + ((k0 + 0) & 3)]
                            : 0.0f;
        bm[c].y = (n16 < 4) ? Tb[((k0 + 1) >> 2) * 64 + n16 * 16 + ((k0 + 1) & 3)]
                            : 0.0f;
      }
      for (int t = wsub * 16; t < wsub * 16 + 16; t += 2) {
        const int row0 = t * 16 + n16;
        const int row1 = (t + 1) * 16 + n16;
        v8f c0 = {0.f, 0.f, 0.f, 0.f, 0.f, 0.f, 0.f, 0.f};
        v8f c1 = {0.f, 0.f, 0.f, 0.f, 0.f, 0.f, 0.f, 0.f};
#pragma unroll
        for (int c = 0; c < 4; ++c) {
          const int m0 = 4 * c + 2 * hi;
          v2f a0, a1;
          a0.x = bufB[row0 * 16 + m0 + 0];
          a0.y = bufB[row0 * 16 + m0 + 1];
          a1.x = bufB[row1 * 16 + m0 + 0];
          a1.y = bufB[row1 * 16 + m0 + 1];
          c0 = wmma_f32(a0, bm[c], c0);
          c1 = wmma_f32(a1, bm[c], c1);
        }
        if (n16 < 4) {
#pragma unroll
          for (int v = 0; v < 8; ++v) {
            bufA[(t * 16 + v + 8 * hi) * 4 + n16] = c0[v];
            bufA[((t + 1) * 16 + v + 8 * hi) * 4 + n16] = c1[v];
          }
        }
      }
    }
    __syncthreads();
  }

  // Amplitude = state with all right dangling bonds at index 0 -> element 0.
  if ((tid & 127) == 0) out[sample] = bufA[0];
}

extern "C" void kernel_launch(void* const* d_in, const int* in_sizes, int n_in,
                              void* d_out, int out_size, void* d_ws,
                              size_t ws_size, hipStream_t stream) {
  const int* x = (const int*)d_in[0];    // [1024, 36] int32
  const float* T = (const float*)d_in[1];  // [6,6,2,4,4,4,4] f32
  float* out = (float*)d_out;              // [1024] f32
  const int batch = in_sizes[0] / (LXR * LYR);
  dim3 grid(batch / 2), block(256);
  peps_amp_kernel<<<grid, block, 0, stream>>>(x, T, out);
}